// SetAbstractionLayer_89515708383360
// MI455X (gfx1250) — compile-verified
//
#include <hip/hip_runtime.h>
#include <math.h>

#define N_PTS   65536
#define S_SMP   2048
#define K_NBR   32
#define RADIUS2 0.0025f
#define C_IN    64
#define C_OUT   128
#define CK1     68          // 67 padded to multiple of 4
#define BN_EPS  1e-5f
#define INF_F   3.402823466e38f
#define MAXC    1024

typedef float v2f __attribute__((ext_vector_type(2)));
typedef float v8f __attribute__((ext_vector_type(8)));

// ---------------- SoA prep ----------------
__global__ void prep_soa(const float* __restrict__ pos, float* __restrict__ px,
                         float* __restrict__ py, float* __restrict__ pz) {
  int i = blockIdx.x * blockDim.x + threadIdx.x;
  if (i < N_PTS) { px[i] = pos[3*i]; py[i] = pos[3*i+1]; pz[i] = pos[3*i+2]; }
}

__global__ void init_misc(int* g_cnt) { if (threadIdx.x == 0) *g_cnt = 0; }

// ---------------- FPS: single persistent workgroup ----------------
__global__ __launch_bounds__(1024) void fps_kernel(
    const float* __restrict__ px, const float* __restrict__ py, const float* __restrict__ pz,
    int* __restrict__ sample_idx, float* __restrict__ pos_s, float* __restrict__ out_pos) {
  __shared__ float rv[1024];
  __shared__ int   ri[1024];
  __shared__ int   s_last;
  const int t = threadIdx.x;
  float md[64];                       // per-lane running min-dist, stays in VGPRs
#pragma unroll
  for (int j = 0; j < 64; ++j) md[j] = INF_F;
  if (t == 0) { s_last = 0; sample_idx[0] = 0; }
  __syncthreads();
  for (int it = 1; it < S_SMP; ++it) {
    const int last = s_last;
    const float lx = px[last], ly = py[last], lz = pz[last];
    float best = -1.0f; int bidx = t;
#pragma unroll
    for (int j = 0; j < 64; ++j) {
      const int i = t + (j << 10);
      const float dx = px[i]-lx, dy = py[i]-ly, dz = pz[i]-lz;
      const float d  = dx*dx + dy*dy + dz*dz;
      const float m  = fminf(md[j], d);
      md[j] = m;
      if (m > best) { best = m; bidx = i; }
    }
    rv[t] = best; ri[t] = bidx;
    __syncthreads();
    for (int off = 512; off > 0; off >>= 1) {
      if (t < off) {
        if (rv[t+off] > rv[t]) { rv[t] = rv[t+off]; ri[t] = ri[t+off]; }
      }
      __syncthreads();
    }
    if (t == 0) { s_last = ri[0]; sample_idx[it] = ri[0]; }
    __syncthreads();
  }
  for (int s = t; s < S_SMP; s += 1024) {
    const int idx = sample_idx[s];
    const float x = px[idx], y = py[idx], z = pz[idx];
    pos_s[3*s]   = x; pos_s[3*s+1]   = y; pos_s[3*s+2]   = z;
    out_pos[3*s] = x; out_pos[3*s+1] = y; out_pos[3*s+2] = z;
  }
}

// ---------------- Ball query: one block per sampled point ----------------
__global__ __launch_bounds__(256) void ballquery_kernel(
    const float* __restrict__ px, const float* __restrict__ py, const float* __restrict__ pz,
    const float* __restrict__ pos_s, int* __restrict__ nbr, int* __restrict__ valid,
    int* __restrict__ g_cnt) {
  __shared__ int   cnt;
  __shared__ float cd[MAXC];
  __shared__ int   ci[MAXC];
  __shared__ float rv[256];
  __shared__ int   rj[256];
  const int s = blockIdx.x, t = threadIdx.x;
  const float sx = pos_s[3*s], sy = pos_s[3*s+1], sz = pos_s[3*s+2];
  if (t == 0) cnt = 0;
  __syncthreads();
  for (int i = t; i < N_PTS; i += 256) {
    __builtin_prefetch(&px[i + 2048], 0, 1);
    const float dx = px[i]-sx, dy = py[i]-sy, dz = pz[i]-sz;
    const float d2 = dx*dx + dy*dy + dz*dz;
    if (d2 <= RADIUS2) {
      const int p = atomicAdd(&cnt, 1);
      if (p < MAXC) { cd[p] = d2; ci[p] = i; }
    }
  }
  __syncthreads();
  const int m = min(cnt, MAXC);
  int nvalid = 0;
  for (int k = 0; k < K_NBR; ++k) {
    float bv = INF_F; int bj = -1;
    for (int j = t; j < m; j += 256) {
      const float v = cd[j];
      if (v < bv || (v == bv && (bj < 0 || ci[j] < ci[bj]))) { bv = v; bj = j; }
    }
    rv[t] = bv; rj[t] = bj;
    __syncthreads();
    for (int off = 128; off > 0; off >>= 1) {
      if (t < off) {
        const float ov = rv[t+off]; const int oj = rj[t+off];
        if (ov < rv[t] || (ov == rv[t] && oj >= 0 && (rj[t] < 0 || ci[oj] < ci[rj[t]]))) {
          rv[t] = ov; rj[t] = oj;
        }
      }
      __syncthreads();
    }
    const int sel = rj[0];
    if (sel >= 0 && rv[0] < INF_F) {
      if (t == 0) { nbr[s*K_NBR+k] = ci[sel]; valid[s*K_NBR+k] = 1; cd[sel] = INF_F; }
      nvalid++;
    } else {
      if (t == 0) { nbr[s*K_NBR+k] = 0; valid[s*K_NBR+k] = 0; }
    }
    __syncthreads();
  }
  if (t == 0) atomicAdd(g_cnt, nvalid);
}

// ---------------- GEMM1: gather + [65536,68]x[68,128] with fp32 WMMA ----------------
__global__ __launch_bounds__(128) void gemm1_kernel(
    const float* __restrict__ h,
    const float* __restrict__ px, const float* __restrict__ py, const float* __restrict__ pz,
    const float* __restrict__ pos_s, const int* __restrict__ nbr, const int* __restrict__ valid,
    const float* __restrict__ W1, const float* __restrict__ b1,
    float* __restrict__ x1, float* __restrict__ psum, float* __restrict__ psq) {
  __shared__ float lw[CK1][C_OUT];        // 34816 B
  __shared__ float la[4][16][CK1];        // 17408 B
  __shared__ float spart_s[128*8];        //  4096 B
  __shared__ float spart_q[128*8];        //  4096 B
  const int t = threadIdx.x;
  const int wave = t >> 5, lane = t & 31;

  for (int i = t; i < CK1*C_OUT; i += 128) {    // stage W1 (row 67 zero-padded)
    const int k = i >> 7, n = i & 127;
    lw[k][n] = (k < 67) ? W1[k*C_OUT + n] : 0.0f;
  }
  const int m_tile = blockIdx.x * 4 + wave;
  {                                              // gather A-tile (16 rows x 68 cols)
    const int row = lane & 15, half = lane >> 4;
    const int gr  = m_tile*16 + row;
    const int s   = gr >> 5;
    const int idx = nbr[gr];
    const float ps0 = pos_s[3*s], ps1 = pos_s[3*s+1], ps2 = pos_s[3*s+2];
    for (int c = half*34; c < half*34 + 34; ++c) {
      float v;
      if      (c < C_IN) v = h[(size_t)idx*C_IN + c];
      else if (c == 64)  v = px[idx] - ps0;
      else if (c == 65)  v = py[idx] - ps1;
      else if (c == 66)  v = pz[idx] - ps2;
      else               v = 0.0f;
      la[wave][row][c] = v;
    }
  }
  __syncthreads();

  v8f acc[8] = {};
  const int row  = lane & 15;
  const int koff = (lane >> 4) << 1;             // ISA 32-bit A layout: lanes16-31 hold K+2
  const int ncol = lane & 15;
#pragma unroll
  for (int k0 = 0; k0 < CK1; k0 += 4) {
    v2f a; a[0] = la[wave][row][k0+koff]; a[1] = la[wave][row][k0+koff+1];
#pragma unroll
    for (int n = 0; n < 8; ++n) {
      v2f b; b[0] = lw[k0+koff][n*16+ncol]; b[1] = lw[k0+koff+1][n*16+ncol];
      acc[n] = __builtin_amdgcn_wmma_f32_16x16x4_f32(
          false, a, false, b, (short)0, acc[n], false, false);
    }
  }

  const int m_base = m_tile*16 + ((lane >> 4) << 3);   // D vgpr i -> row i (+8 for hi lanes)
  float wv[8];
#pragma unroll
  for (int i = 0; i < 8; ++i) wv[i] = valid[m_base + i] ? 1.0f : 0.0f;
#pragma unroll
  for (int n = 0; n < 8; ++n) {
    const int col = n*16 + ncol;
    const float bias = b1[col];
    float ls = 0.0f, lq = 0.0f;
#pragma unroll
    for (int i = 0; i < 8; ++i) {
      const float x = acc[n][i] + bias;
      x1[(size_t)(m_base + i)*C_OUT + col] = x;
      ls += x * wv[i]; lq += x * x * wv[i];
    }
    spart_s[t*8+n] = ls; spart_q[t*8+n] = lq;
  }
  __syncthreads();
  {                                              // deterministic per-block channel reduce
    const int c = t, n = c >> 4, nc = c & 15;
    float ss = 0.0f, sq = 0.0f;
#pragma unroll
    for (int w = 0; w < 4; ++w)
#pragma unroll
      for (int hh = 0; hh < 2; ++hh) {
        const int src = w*32 + hh*16 + nc;
        ss += spart_s[src*8+n]; sq += spart_q[src*8+n];
      }
    psum[(size_t)blockIdx.x*C_OUT + c] = ss;
    psq [(size_t)blockIdx.x*C_OUT + c] = sq;
  }
}

// ---------------- BN finalize (deterministic fixed-order reduce) ----------------
__global__ void bn_finalize_kernel(const float* __restrict__ psum, const float* __restrict__ psq,
                                   int nblocks, const int* __restrict__ g_cnt,
                                   const float* __restrict__ gamma, const float* __restrict__ beta,
                                   float* __restrict__ A, float* __restrict__ B) {
  const int c = threadIdx.x;                     // 128 threads
  float s = 0.0f, q = 0.0f;
  for (int b = 0; b < nblocks; ++b) { s += psum[(size_t)b*C_OUT+c]; q += psq[(size_t)b*C_OUT+c]; }
  const float cnt  = (float)(*g_cnt);
  const float mean = s / cnt;
  const float var  = q / cnt - mean*mean;
  const float inv  = rsqrtf(var + BN_EPS);
  const float a    = gamma[c] * inv;
  A[c] = a;
  B[c] = beta[c] - mean * a;
}

// ---------------- GEMM2: BN+ReLU fused load, [65536,128]x[128,128] WMMA ----------------
__global__ __launch_bounds__(256) void gemm2_kernel(
    const float* __restrict__ x1, const float* __restrict__ bnA, const float* __restrict__ bnB,
    const int* __restrict__ valid, const float* __restrict__ W2, const float* __restrict__ b2,
    float* __restrict__ x2, float* __restrict__ psum, float* __restrict__ psq) {
  extern __shared__ float smem[];
  float* lw      = smem;                    // 128*128
  float* la      = lw + C_OUT*C_OUT;        // 8*16*128
  float* spart_s = la + 8*16*C_OUT;         // 256*8
  float* spart_q = spart_s + 256*8;         // 256*8
  const int t = threadIdx.x;
  const int wave = t >> 5, lane = t & 31;

  for (int i = t; i < C_OUT*C_OUT; i += 256) lw[i] = W2[i];    // lw[k*128+n]
  const int m_tile = blockIdx.x * 8 + wave;
  {
    const int row = lane & 15, half = lane >> 4;
    const int gr  = m_tile*16 + row;
    for (int c = half*64; c < half*64 + 64; ++c) {
      float v = x1[(size_t)gr*C_OUT + c];
      v = fmaxf(bnA[c]*v + bnB[c], 0.0f);                      // fused BN1 + ReLU
      la[(wave*16+row)*C_OUT + c] = v;
    }
  }
  __syncthreads();

  v8f acc[8] = {};
  const int row = lane & 15, koff = (lane >> 4) << 1, ncol = lane & 15;
  for (int k0 = 0; k0 < C_OUT; k0 += 4) {
    v2f a; a[0] = la[(wave*16+row)*C_OUT + k0+koff];
           a[1] = la[(wave*16+row)*C_OUT + k0+koff+1];
#pragma unroll
    for (int n = 0; n < 8; ++n) {
      v2f b; b[0] = lw[(k0+koff)*C_OUT + n*16+ncol];
             b[1] = lw[(k0+koff+1)*C_OUT + n*16+ncol];
      acc[n] = __builtin_amdgcn_wmma_f32_16x16x4_f32(
          false, a, false, b, (short)0, acc[n], false, false);
    }
  }

  const int m_base = m_tile*16 + ((lane >> 4) << 3);
  float wv[8];
#pragma unroll
  for (int i = 0; i < 8; ++i) wv[i] = valid[m_base + i] ? 1.0f : 0.0f;
#pragma unroll
  for (int n = 0; n < 8; ++n) {
    const int col = n*16 + ncol;
    const float bias = b2[col];
    float ls = 0.0f, lq = 0.0f;
#pragma unroll
    for (int i = 0; i < 8; ++i) {
      const float x = acc[n][i] + bias;
      x2[(size_t)(m_base + i)*C_OUT + col] = x;
      ls += x * wv[i]; lq += x * x * wv[i];
    }
    spart_s[t*8+n] = ls; spart_q[t*8+n] = lq;
  }
  __syncthreads();
  if (t < 128) {
    const int n = t >> 4, nc = t & 15;
    float ss = 0.0f, sq = 0.0f;
#pragma unroll
    for (int w = 0; w < 8; ++w)
#pragma unroll
      for (int hh = 0; hh < 2; ++hh) {
        const int src = w*32 + hh*16 + nc;
        ss += spart_s[src*8+n]; sq += spart_q[src*8+n];
      }
    psum[(size_t)blockIdx.x*C_OUT + t] = ss;
    psq [(size_t)blockIdx.x*C_OUT + t] = sq;
  }
}

// ---------------- Max-pool over K with BN2+ReLU fused ----------------
__global__ __launch_bounds__(128) void maxpool_kernel(
    const float* __restrict__ x2, const float* __restrict__ bnA, const float* __restrict__ bnB,
    const int* __restrict__ valid, float* __restrict__ out) {
  const int s = blockIdx.x, c = threadIdx.x;
  const float a = bnA[c], b = bnB[c];
  float best = -INF_F; int any = 0;
  for (int k = 0; k < K_NBR; ++k) {
    if (valid[s*K_NBR + k]) {
      const float x = fmaxf(a * x2[(size_t)(s*K_NBR + k)*C_OUT + c] + b, 0.0f);
      best = fmaxf(best, x); any = 1;
    }
  }
  out[(size_t)s*C_OUT + c] = any ? best : 0.0f;
}

extern "C" void kernel_launch(void* const* d_in, const int* in_sizes, int n_in,
                              void* d_out, int out_size, void* d_ws, size_t ws_size,
                              hipStream_t stream) {
  (void)in_sizes; (void)n_in; (void)out_size; (void)ws_size;
  const float* h   = (const float*)d_in[0];
  const float* pos = (const float*)d_in[1];
  const float* W1  = (const float*)d_in[2];
  const float* b1  = (const float*)d_in[3];
  const float* g1  = (const float*)d_in[4];
  const float* be1 = (const float*)d_in[5];
  const float* W2  = (const float*)d_in[6];
  const float* b2  = (const float*)d_in[7];
  const float* g2  = (const float*)d_in[8];
  const float* be2 = (const float*)d_in[9];
  float* out = (float*)d_out;

  char* ws = (char*)d_ws;
  float* px         = (float*)(ws + 0);
  float* py         = (float*)(ws + 262144);
  float* pz         = (float*)(ws + 524288);
  int*   sample_idx = (int*)  (ws + 786432);
  float* pos_s      = (float*)(ws + 794624);
  int*   nbr        = (int*)  (ws + 819200);
  int*   valid      = (int*)  (ws + 1081344);
  int*   g_cnt      = (int*)  (ws + 1343488);
  float* bnA1       = (float*)(ws + 1343744);
  float* bnB1       = (float*)(ws + 1344256);
  float* bnA2       = (float*)(ws + 1344768);
  float* bnB2       = (float*)(ws + 1345280);
  float* psum1      = (float*)(ws + 1572864);
  float* psq1       = (float*)(ws + 2097152);
  float* psum2      = (float*)(ws + 2621440);
  float* psq2       = (float*)(ws + 2883584);
  float* x1         = (float*)(ws + 3145728);
  float* x2         = (float*)(ws + 36700160);

  prep_soa<<<256, 256, 0, stream>>>(pos, px, py, pz);
  init_misc<<<1, 64, 0, stream>>>(g_cnt);
  fps_kernel<<<1, 1024, 0, stream>>>(px, py, pz, sample_idx, pos_s,
                                     out + (size_t)S_SMP * C_OUT);
  ballquery_kernel<<<S_SMP, 256, 0, stream>>>(px, py, pz, pos_s, nbr, valid, g_cnt);
  gemm1_kernel<<<1024, 128, 0, stream>>>(h, px, py, pz, pos_s, nbr, valid, W1, b1,
                                         x1, psum1, psq1);
  bn_finalize_kernel<<<1, 128, 0, stream>>>(psum1, psq1, 1024, g_cnt, g1, be1, bnA1, bnB1);
  const size_t dynbytes = (size_t)(C_OUT*C_OUT + 8*16*C_OUT + 256*8*2) * sizeof(float); // 144 KB
  gemm2_kernel<<<512, 256, dynbytes, stream>>>(x1, bnA1, bnB1, valid, W2, b2,
                                               x2, psum2, psq2);
  bn_finalize_kernel<<<1, 128, 0, stream>>>(psum2, psq2, 512, g_cnt, g2, be2, bnA2, bnB2);
  maxpool_kernel<<<S_SMP, 128, 0, stream>>>(x2, bnA2, bnB2, valid, out);
}